// HyperNet_13013750907195
// MI455X (gfx1250) — compile-verified
//
#include <hip/hip_runtime.h>

#define T_TOTAL 35968706LL
#define BATCH 4

typedef __attribute__((ext_vector_type(2))) float v2f;
typedef __attribute__((ext_vector_type(8))) float v8f;

// ---------------------------------------------------------------------------
// Direct strided conv: kernel 4x4, stride 2, pad 1, optional LeakyReLU(0.01)
// ---------------------------------------------------------------------------
__global__ void conv_s2_kernel(const float* __restrict__ in,
                               const float* __restrict__ w,
                               const float* __restrict__ bias,
                               float* __restrict__ out,
                               int Cin, int Cout, int Hin, int Win,
                               int Hout, int Wout, int do_lrelu)
{
    int idx = blockIdx.x * blockDim.x + threadIdx.x;
    int total = BATCH * Cout * Hout * Wout;
    if (idx >= total) return;
    int wo = idx % Wout; int t = idx / Wout;
    int ho = t % Hout;   t /= Hout;
    int co = t % Cout;   int n = t / Cout;

    float acc = bias[co];
    int hbase = ho * 2 - 1;
    int wbase = wo * 2 - 1;
    const float* wp = w + (size_t)co * Cin * 16;
    const float* ip = in + (size_t)n * Cin * Hin * Win;
    for (int ci = 0; ci < Cin; ++ci) {
        const float* ipc = ip + (size_t)ci * Hin * Win;
        const float* wpc = wp + ci * 16;
        #pragma unroll
        for (int kh = 0; kh < 4; ++kh) {
            int h = hbase + kh;
            if ((unsigned)h >= (unsigned)Hin) continue;
            #pragma unroll
            for (int kw = 0; kw < 4; ++kw) {
                int x = wbase + kw;
                if ((unsigned)x >= (unsigned)Win) continue;
                acc = fmaf(ipc[h * Win + x], wpc[kh * 4 + kw], acc);
            }
        }
    }
    if (do_lrelu && acc < 0.0f) acc *= 0.01f;
    out[idx] = acc;
}

// ---------------------------------------------------------------------------
// Head: adaptive-avg-pool(1) -> fc1+lrelu -> fc2 -> VQ argmin -> loss, q
// Single block of 256 threads; everything is tiny ([4,64] -> [4,16]).
// ---------------------------------------------------------------------------
__global__ void head_kernel(const float* __restrict__ conv5,   // [4,64,7,7]
                            const float* __restrict__ fc1_w,   // [16,64]
                            const float* __restrict__ fc1_b,   // [16]
                            const float* __restrict__ fc2_w,   // [16,16]
                            const float* __restrict__ fc2_b,   // [16]
                            const float* __restrict__ emb,     // [4,16]
                            float* __restrict__ qbuf,          // [4*16] out
                            float* __restrict__ loss_out)      // [1] out
{
    __shared__ float pooled[256];   // [4][64]
    __shared__ float x1[64];        // [4][16]
    __shared__ float evec[64];      // [4][16]
    __shared__ float qs[64];        // [4][16]
    __shared__ float diff2[64];
    int t = threadIdx.x;

    // pool: mean over 7x7
    {
        const float* p = conv5 + (size_t)t * 49;
        float s = 0.0f;
        for (int i = 0; i < 49; ++i) s += p[i];
        pooled[t] = s * (1.0f / 49.0f);
    }
    __syncthreads();

    if (t < 64) {   // fc1 + lrelu
        int n = t >> 4, j = t & 15;
        float s = fc1_b[j];
        for (int c = 0; c < 64; ++c) s = fmaf(pooled[n * 64 + c], fc1_w[j * 64 + c], s);
        x1[t] = (s >= 0.0f) ? s : 0.01f * s;
    }
    __syncthreads();

    if (t < 64) {   // fc2 -> e
        int n = t >> 4, i = t & 15;
        float s = fc2_b[i];
        for (int j = 0; j < 16; ++j) s = fmaf(x1[n * 16 + j], fc2_w[i * 16 + j], s);
        evec[t] = s;
    }
    __syncthreads();

    if (t < BATCH) {    // VQ nearest codebook (first-min tie break like argmin)
        float best = 3.402823466e+38f; int bi = 0;
        for (int k = 0; k < 4; ++k) {
            float d = 0.0f;
            for (int i = 0; i < 16; ++i) {
                float df = evec[t * 16 + i] - emb[k * 16 + i];
                d = fmaf(df, df, d);
            }
            if (d < best) { best = d; bi = k; }
        }
        for (int i = 0; i < 16; ++i) {
            float q = emb[bi * 16 + i];
            qs[t * 16 + i] = q;
            float df = q - evec[t * 16 + i];
            diff2[t * 16 + i] = df * df;
        }
    }
    __syncthreads();

    if (t < 64) qbuf[t] = qs[t];            // q_st forward value == q
    if (t == 0) {
        float s = 0.0f;
        for (int i = 0; i < 64; ++i) s += diff2[i];
        // q_lat + 0.25*e_lat; both equal mean((q-e)^2) in the forward pass
        *loss_out = 1.25f * (s * (1.0f / 64.0f));
    }
}

// ---------------------------------------------------------------------------
// flat[4, T] = q[4,16] @ W[16, T] via V_WMMA_F32_16X16X4_F32.
//
// Layouts (ISA 7.12.2, 32-bit ops):
//   A 16x4 slice of q (rows >= 4 zero): lanes 0-15 carry K={0,1}, lanes
//   16-31 carry K={2,3}; VGPR0/1 = the two K values.
//   B 4x16 slice of W: mirrored K striping, N = lane&15.
//   D 16x16: acc[r] = row M=r at N=lane for lanes 0-15, so the 4 real batch
//   rows store as coalesced 64B segments from lanes 0-15.
//
// Codegen strategy:
//   * The 8 W-row bases (W + {0,1,4,5,8,9,12,13}*T) and 4 out-row bases are
//     KERNEL ARGUMENTS -> opaque uniform SGPR pairs, so every load/store is
//     the SADDR form sharing ONE 32-bit vector offset (scale_offset), with
//     the pair's second tile reached via the +64B instruction offset.
//     The lane-dependent +2*kh*T row shift lives in that vector offset
//     (max 3*T ~ 108M elements < 2^31).
//   * Main loop covers only FULL 32-column pairs -> zero predication; 16
//     clause-grouped loads, 8 chained WMMAs with staged partial
//     s_wait_loadcnt, 8 coalesced stores.
//   * 2-column tail (T % 32 == 2) handled scalar by wave 0.
//
// Roofline: 2.30 GB W read + 0.58 GB store @ 23.3 TB/s => ~124 us floor;
// 4.6 GFLOP is negligible, so this kernel is pure stream + matrix pipe.
// ---------------------------------------------------------------------------
__global__ void hyper_gemm_wmma_kernel(
        const float* __restrict__ Wr0,  const float* __restrict__ Wr1,
        const float* __restrict__ Wr4,  const float* __restrict__ Wr5,
        const float* __restrict__ Wr8,  const float* __restrict__ Wr9,
        const float* __restrict__ Wr12, const float* __restrict__ Wr13,
        float* __restrict__ out0, float* __restrict__ out1,
        float* __restrict__ out2, float* __restrict__ out3,
        const float* __restrict__ qbuf)
{
    const long long T = T_TOTAL;
    const long long NPAIR = T >> 5;            // 1,124,022 full 32-col pairs
    int lane = threadIdx.x & 31;
    int m    = lane & 15;
    int kh   = lane >> 4;                      // 0: lanes 0-15, 1: lanes 16-31

    // Hoisted A operand (q zero-padded to 16 rows) for all four K-slices.
    v2f a[4];
    #pragma unroll
    for (int s = 0; s < 4; ++s) {
        int k0 = 4 * s + 2 * kh;
        a[s].x = (m < BATCH) ? qbuf[m * 16 + k0]     : 0.0f;
        a[s].y = (m < BATCH) ? qbuf[m * 16 + k0 + 1] : 0.0f;
    }

    long long wave   = (long long)((blockIdx.x * blockDim.x + threadIdx.x) >> 5);
    long long nwaves = ((long long)gridDim.x * blockDim.x) >> 5;

    const unsigned khoff = (unsigned)(2 * kh) * (unsigned)T_TOTAL;  // per-lane row shift

    for (long long p = wave; p < NPAIR; p += nwaves) {
        unsigned c0 = (unsigned)(p << 5) + (unsigned)m;   // first tile's column
        unsigned eo = khoff + c0;                         // element offset, < 2^31

        // 16 unpredicated loads: 8 rows x 2 tiles (tile 2 at +16 elements).
        float x00 = Wr0 [eo], x01 = Wr1 [eo];
        float x10 = Wr4 [eo], x11 = Wr5 [eo];
        float x20 = Wr8 [eo], x21 = Wr9 [eo];
        float x30 = Wr12[eo], x31 = Wr13[eo];
        float y00 = Wr0 [eo + 16], y01 = Wr1 [eo + 16];
        float y10 = Wr4 [eo + 16], y11 = Wr5 [eo + 16];
        float y20 = Wr8 [eo + 16], y21 = Wr9 [eo + 16];
        float y30 = Wr12[eo + 16], y31 = Wr13[eo + 16];

        v8f acc0 = {};
        v8f acc1 = {};
        { v2f b; b.x = x00; b.y = x01;
          acc0 = __builtin_amdgcn_wmma_f32_16x16x4_f32(false, a[0], false, b, (short)0, acc0, false, false); }
        { v2f b; b.x = y00; b.y = y01;
          acc1 = __builtin_amdgcn_wmma_f32_16x16x4_f32(false, a[0], false, b, (short)0, acc1, false, false); }
        { v2f b; b.x = x10; b.y = x11;
          acc0 = __builtin_amdgcn_wmma_f32_16x16x4_f32(false, a[1], false, b, (short)0, acc0, false, false); }
        { v2f b; b.x = y10; b.y = y11;
          acc1 = __builtin_amdgcn_wmma_f32_16x16x4_f32(false, a[1], false, b, (short)0, acc1, false, false); }
        { v2f b; b.x = x20; b.y = x21;
          acc0 = __builtin_amdgcn_wmma_f32_16x16x4_f32(false, a[2], false, b, (short)0, acc0, false, false); }
        { v2f b; b.x = y20; b.y = y21;
          acc1 = __builtin_amdgcn_wmma_f32_16x16x4_f32(false, a[2], false, b, (short)0, acc1, false, false); }
        { v2f b; b.x = x30; b.y = x31;
          acc0 = __builtin_amdgcn_wmma_f32_16x16x4_f32(false, a[3], false, b, (short)0, acc0, false, false); }
        { v2f b; b.x = y30; b.y = y31;
          acc1 = __builtin_amdgcn_wmma_f32_16x16x4_f32(false, a[3], false, b, (short)0, acc1, false, false); }

        if (lane < 16) {    // rows 0..3 of both tiles, coalesced 64B stores
            out0[c0] = acc0[0]; out1[c0] = acc0[1];
            out2[c0] = acc0[2]; out3[c0] = acc0[3];
            out0[c0 + 16] = acc1[0]; out1[c0 + 16] = acc1[1];
            out2[c0 + 16] = acc1[2]; out3[c0 + 16] = acc1[3];
        }
    }

    // Scalar tail: T % 32 == 2 leftover columns, done by wave 0.
    if (wave == 0) {
        int rem = (int)(T & 31);
        if (lane < rem) {
            long long c = (NPAIR << 5) + lane;
            float s0 = 0.f, s1 = 0.f, s2 = 0.f, s3 = 0.f;
            for (int k = 0; k < 16; ++k) {
                float wv = Wr0[(long long)k * T + c];
                s0 = fmaf(qbuf[0 * 16 + k], wv, s0);
                s1 = fmaf(qbuf[1 * 16 + k], wv, s1);
                s2 = fmaf(qbuf[2 * 16 + k], wv, s2);
                s3 = fmaf(qbuf[3 * 16 + k], wv, s3);
            }
            out0[c] = s0; out1[c] = s1; out2[c] = s2; out3[c] = s3;
        }
    }
}

// ---------------------------------------------------------------------------
extern "C" void kernel_launch(void* const* d_in, const int* in_sizes, int n_in,
                              void* d_out, int out_size, void* d_ws, size_t ws_size,
                              hipStream_t stream) {
    const float* rgb   = (const float*)d_in[0];
    const float* cw1   = (const float*)d_in[1];
    const float* cb1   = (const float*)d_in[2];
    const float* cw2   = (const float*)d_in[3];
    const float* cb2   = (const float*)d_in[4];
    const float* cw3   = (const float*)d_in[5];
    const float* cb3   = (const float*)d_in[6];
    const float* cw4   = (const float*)d_in[7];
    const float* cb4   = (const float*)d_in[8];
    const float* cw5   = (const float*)d_in[9];
    const float* cb5   = (const float*)d_in[10];
    const float* fc1_w = (const float*)d_in[11];
    const float* fc1_b = (const float*)d_in[12];
    const float* fc2_w = (const float*)d_in[13];
    const float* fc2_b = (const float*)d_in[14];
    const float* emb   = (const float*)d_in[15];
    const float* W_all = (const float*)d_in[16];

    float* out = (float*)d_out;
    float* ws  = (float*)d_ws;

    float* a1   = ws;                    // [4,4,112,112] = 200704
    float* a2   = a1 + 200704;           // [4,8,56,56]   = 100352
    float* a3   = a2 + 100352;           // [4,16,28,28]  =  50176
    float* a4   = a3 + 50176;            // [4,32,14,14]  =  25088
    float* a5   = a4 + 25088;            // [4,64,7,7]    =  12544
    float* qbuf = a5 + 12544;            // [4,16]

    int tot;
    tot = BATCH * 4 * 112 * 112;
    conv_s2_kernel<<<(tot + 255) / 256, 256, 0, stream>>>(rgb, cw1, cb1, a1, 3,  4, 224, 224, 112, 112, 1);
    tot = BATCH * 8 * 56 * 56;
    conv_s2_kernel<<<(tot + 255) / 256, 256, 0, stream>>>(a1,  cw2, cb2, a2, 4,  8, 112, 112,  56,  56, 1);
    tot = BATCH * 16 * 28 * 28;
    conv_s2_kernel<<<(tot + 255) / 256, 256, 0, stream>>>(a2,  cw3, cb3, a3, 8, 16,  56,  56,  28,  28, 1);
    tot = BATCH * 32 * 14 * 14;
    conv_s2_kernel<<<(tot + 255) / 256, 256, 0, stream>>>(a3,  cw4, cb4, a4, 16, 32, 28,  28,  14,  14, 1);
    tot = BATCH * 64 * 7 * 7;
    conv_s2_kernel<<<(tot + 255) / 256, 256, 0, stream>>>(a4,  cw5, cb5, a5, 32, 64, 14,  14,   7,   7, 0);

    head_kernel<<<1, 256, 0, stream>>>(a5, fc1_w, fc1_b, fc2_w, fc2_b, emb,
                                       qbuf, out + 4 * T_TOTAL);

    // Precompute uniform row-base pointers on the host so they land in SGPRs.
    const long long T = T_TOTAL;
    hyper_gemm_wmma_kernel<<<4096, 256, 0, stream>>>(
        W_all,          W_all + 1 * T,
        W_all + 4 * T,  W_all + 5 * T,
        W_all + 8 * T,  W_all + 9 * T,
        W_all + 12 * T, W_all + 13 * T,
        out,            out + 1 * T,
        out + 2 * T,    out + 3 * T,
        qbuf);
}